// Net_22101901705839
// MI455X (gfx1250) — compile-verified
//
#include <hip/hip_runtime.h>
#include <hip/hip_bf16.h>

typedef float v2f __attribute__((ext_vector_type(2)));
typedef float v8f __attribute__((ext_vector_type(8)));

// ---------------------------------------------------------------------------
// g1[N,16] = feat[N,64] @ W1[64,16]   using V_WMMA_F32_16X16X4_F32 (fp32 WMMA)
// One wave32 computes one 16x16 output tile (16 rows of g1).
// A 16x4 layout: lanes 0-15 -> M=lane, K={0,1}; lanes 16-31 -> M=lane-16, K={2,3}
// B 4x16 layout: lanes 0-15 -> N=lane, K={0,1}; lanes 16-31 -> N=lane-16, K={2,3}
// C/D 16x16:     vgpr i, lanes 0-15 -> M=i, N=lane; lanes 16-31 -> M=i+8
// ---------------------------------------------------------------------------
__global__ __launch_bounds__(256) void proj1_wmma(const float* __restrict__ feat,
                                                  const float* __restrict__ W1,
                                                  float* __restrict__ g1,
                                                  int ntiles) {
    const int wave = threadIdx.x >> 5;               // 8 waves per block
    const int tile = blockIdx.x * 8 + wave;          // 16 rows per tile
    if (tile >= ntiles) return;
    const int lane = threadIdx.x & 31;
    const int m    = lane & 15;                      // row (A) / col (B,C)
    const int kh   = (lane >> 4) * 2;                // K sub-offset: 0 or 2
    const int row0 = tile * 16;

    const float* __restrict__ arow = feat + (size_t)(row0 + m) * 64;

    v8f c = {};
#pragma unroll
    for (int kk = 0; kk < 64; kk += 4) {
        v2f a, b;
        a.x = arow[kk + kh + 0];
        a.y = arow[kk + kh + 1];
        b.x = W1[(kk + kh + 0) * 16 + m];
        b.y = W1[(kk + kh + 1) * 16 + m];
        c = __builtin_amdgcn_wmma_f32_16x16x4_f32(
                /*neg_a=*/false, a, /*neg_b=*/false, b,
                /*c_mod=*/(short)0, c, /*reuse_a=*/false, /*reuse_b=*/false);
    }

    const int rbase = row0 + ((lane >> 4) * 8);
#pragma unroll
    for (int i = 0; i < 8; ++i) {
        g1[(size_t)(rbase + i) * 16 + m] = c[i];
    }
}

// ---------------------------------------------------------------------------
// Edge aggregation, 16 channels: agg1[dst] += (w+1) * g1[src]
// 16 threads per edge -> coalesced 64B gathers; g1 (6.4MB) is L2-resident.
// ---------------------------------------------------------------------------
__global__ __launch_bounds__(256) void edge_scatter16(const int* __restrict__ src,
                                                      const int* __restrict__ dst,
                                                      const float* __restrict__ ew,
                                                      const float* __restrict__ g1,
                                                      float* __restrict__ agg1,
                                                      int nEdges) {
    long long t = (long long)blockIdx.x * blockDim.x + threadIdx.x;
    int e = (int)(t >> 4);
    int c = (int)(t & 15);
    if (e >= nEdges) return;
    int   s = src[e];
    int   d = dst[e];
    float w = ew[e] + 1.0f;
    float v = w * g1[(size_t)s * 16 + c];
#if defined(__HIP_DEVICE_COMPILE__)
    unsafeAtomicAdd(&agg1[(size_t)d * 16 + c], v);   // hardware global_atomic_add_f32
#else
    atomicAdd(&agg1[(size_t)d * 16 + c], v);
#endif
}

// ---------------------------------------------------------------------------
// h = relu(g1*(sw+1) + agg1 + b1);  g2[N,2] = h @ W2  (fold layer-2 projection)
// ---------------------------------------------------------------------------
__global__ __launch_bounds__(256) void combine1(const float* __restrict__ g1,
                                                const float* __restrict__ agg1,
                                                const float* __restrict__ sw,
                                                const float* __restrict__ b1,
                                                const float* __restrict__ W2,
                                                float* __restrict__ g2,
                                                int n) {
    int i = blockIdx.x * blockDim.x + threadIdx.x;
    if (i >= n) return;
    float s  = sw[i] + 1.0f;
    float o0 = 0.0f, o1 = 0.0f;
#pragma unroll
    for (int c = 0; c < 16; ++c) {
        float h = fmaf(g1[(size_t)i * 16 + c], s, agg1[(size_t)i * 16 + c]) + b1[c];
        h = fmaxf(h, 0.0f);
        o0 = fmaf(h, W2[c * 2 + 0], o0);
        o1 = fmaf(h, W2[c * 2 + 1], o1);
    }
    g2[(size_t)i * 2 + 0] = o0;
    g2[(size_t)i * 2 + 1] = o1;
}

// ---------------------------------------------------------------------------
// Edge aggregation, 2 channels: agg2[dst] += (w+1) * g2[src]
// ---------------------------------------------------------------------------
__global__ __launch_bounds__(256) void edge_scatter2(const int* __restrict__ src,
                                                     const int* __restrict__ dst,
                                                     const float* __restrict__ ew,
                                                     const float* __restrict__ g2,
                                                     float* __restrict__ agg2,
                                                     int nEdges) {
    int e = blockIdx.x * blockDim.x + threadIdx.x;
    if (e >= nEdges) return;
    int   s = src[e];
    int   d = dst[e];
    float w = ew[e] + 1.0f;
    const float2* gp = (const float2*)(g2 + (size_t)s * 2);
    float2 g = *gp;
#if defined(__HIP_DEVICE_COMPILE__)
    unsafeAtomicAdd(&agg2[(size_t)d * 2 + 0], w * g.x);
    unsafeAtomicAdd(&agg2[(size_t)d * 2 + 1], w * g.y);
#else
    atomicAdd(&agg2[(size_t)d * 2 + 0], w * g.x);
    atomicAdd(&agg2[(size_t)d * 2 + 1], w * g.y);
#endif
}

// ---------------------------------------------------------------------------
// out[N,2] = g2*(sw+1) + agg2 + b2
// ---------------------------------------------------------------------------
__global__ __launch_bounds__(256) void combine2(const float* __restrict__ g2,
                                                const float* __restrict__ agg2,
                                                const float* __restrict__ sw,
                                                const float* __restrict__ b2,
                                                float* __restrict__ out,
                                                int n) {
    int i = blockIdx.x * blockDim.x + threadIdx.x;
    if (i >= n) return;
    float s = sw[i] + 1.0f;
    out[(size_t)i * 2 + 0] = fmaf(g2[(size_t)i * 2 + 0], s, agg2[(size_t)i * 2 + 0]) + b2[0];
    out[(size_t)i * 2 + 1] = fmaf(g2[(size_t)i * 2 + 1], s, agg2[(size_t)i * 2 + 1]) + b2[1];
}

extern "C" void kernel_launch(void* const* d_in, const int* in_sizes, int n_in,
                              void* d_out, int out_size, void* d_ws, size_t ws_size,
                              hipStream_t stream) {
    const float* feat = (const float*)d_in[0];
    const int*   src  = (const int*)d_in[1];
    const int*   dst  = (const int*)d_in[2];
    const float* ew   = (const float*)d_in[3];
    const float* sw   = (const float*)d_in[4];
    const float* W1   = (const float*)d_in[5];
    const float* b1   = (const float*)d_in[6];
    const float* W2   = (const float*)d_in[7];
    const float* b2   = (const float*)d_in[8];
    float* out = (float*)d_out;

    const int N = in_sizes[4];   // self_weight: N_NODES
    const int E = in_sizes[1];   // src: N_EDGES

    // Workspace layout (floats): g1[N*16] | agg1[N*16] | g2[N*2] | agg2[N*2]
    float* g1   = (float*)d_ws;
    float* agg1 = g1   + (size_t)N * 16;
    float* g2   = agg1 + (size_t)N * 16;
    float* agg2 = g2   + (size_t)N * 2;

    // Zero the atomic accumulation buffers every call (graph-capture safe).
    hipMemsetAsync(agg1, 0, (size_t)N * 16 * sizeof(float), stream);
    hipMemsetAsync(agg2, 0, (size_t)N * 2 * sizeof(float), stream);

    const int ntiles = (N + 15) / 16;                       // 6250 tiles (N is 16-aligned)
    proj1_wmma<<<(ntiles + 7) / 8, 256, 0, stream>>>(feat, W1, g1, ntiles);

    long long t16 = (long long)E * 16;
    edge_scatter16<<<(int)((t16 + 255) / 256), 256, 0, stream>>>(src, dst, ew, g1, agg1, E);

    combine1<<<(N + 255) / 256, 256, 0, stream>>>(g1, agg1, sw, b1, W2, g2, N);

    edge_scatter2<<<(E + 255) / 256, 256, 0, stream>>>(src, dst, ew, g2, agg2, E);

    combine2<<<(N + 255) / 256, 256, 0, stream>>>(g2, agg2, sw, b2, out, N);
}